// Distance_37022618091794
// MI455X (gfx1250) — compile-verified
//
#include <hip/hip_runtime.h>
#include <math.h>

typedef __attribute__((ext_vector_type(2))) float v2f;
typedef __attribute__((ext_vector_type(4))) float f4;
typedef __attribute__((ext_vector_type(8))) float v8f;

// ---------------------------------------------------------------------------
// Kernel 1: bandwidth-bound streaming copy (adj_mats -> out_adj, ew -> out_ew)
// B128 non-temporal loads/stores: 2.15 GB of traffic, ~92 us at 23.3 TB/s.
// NT hint avoids polluting the 192 MB L2 with stream-once data.
// ---------------------------------------------------------------------------
__global__ __launch_bounds__(256) void stream_copy_kernel(
    const f4* __restrict__ src, f4* __restrict__ dst, long n4) {
  long i = (long)blockIdx.x * blockDim.x + threadIdx.x;
  long stride = (long)gridDim.x * blockDim.x;
  for (; i < n4; i += stride) {
    f4 v = __builtin_nontemporal_load(src + i);
    __builtin_nontemporal_store(v, dst + i);
  }
}

// ---------------------------------------------------------------------------
// Kernel 2: per-batch distance mask + row/col scatter.
// d2_i = diag((X - c)(X - c)^T), computed with V_WMMA_F32_16X16X4_F32 in
// full fp32.  One workgroup (8 wave32) per batch; each wave owns 16-node
// tiles.  A-matrix fp32 16x4 layout (ISA 7.12.2): lane L holds node L%16,
// VGPR v holds K = 2*(L/16) + v; the 4x16 B layout is the mirror with
// N = L%16, so identical per-lane diff pairs serve as both A and B.
// Diagonal of the 16x16 f32 C/D tile: node i in 0..7 -> lane i, VGPR i;
// node i in 8..15 -> lane i+16, VGPR i-8.
// ---------------------------------------------------------------------------
__global__ __launch_bounds__(256) void dist_scatter_kernel(
    const float* __restrict__ nodes, const long long* __restrict__ num_nodes,
    float* __restrict__ out_adj, int N, int F) {
  int b = blockIdx.x;
  int nn = (int)num_nodes[b];

  __shared__ float csh[64];  // current-node feature vector (F <= 64)
  const float* nb = nodes + (size_t)b * N * F;
  for (int t = threadIdx.x; t < F; t += blockDim.x)
    csh[t] = nb[(size_t)nn * F + t];
  __syncthreads();

  int lane = threadIdx.x & 31;
  int wave = threadIdx.x >> 5;
  int nwaves = blockDim.x >> 5;
  int lo = lane & 15;   // node-in-tile (A) / column (B)
  int hi = lane >> 4;   // K-half selector
  int ntiles = N >> 4;
  int kchunks = F >> 2;

  for (int t = wave; t < ntiles; t += nwaves) {
    int node = t * 16 + lo;
    const float* xp = nb + (size_t)node * F;
    v8f acc = {0.f, 0.f, 0.f, 0.f, 0.f, 0.f, 0.f, 0.f};

    for (int kc = 0; kc < kchunks; ++kc) {
      int k0 = 4 * kc + 2 * hi;
      v2f a;
      a.x = xp[k0] - csh[k0];
      a.y = xp[k0 + 1] - csh[k0 + 1];
#if __has_builtin(__builtin_amdgcn_wmma_f32_16x16x4_f32)
      acc = __builtin_amdgcn_wmma_f32_16x16x4_f32(
          false, a, false, a, (short)0, acc, false, false);
#else
      // Fallback (should not be taken on gfx1250 per probe): accumulate the
      // diagonal contribution directly into element 0.
      acc[0] += a.x * a.x + a.y * a.y;
#endif
    }

    // Extract this lane's diagonal element (if it owns one).
    bool active = (lane < 8) || (lane >= 24);
    int r = (lane < 8) ? lane : (lane - 24);  // VGPR index of diag element
    int i = (lane < 8) ? lane : (lane - 16);  // node-in-tile of diag element
    float d2 = acc[0];
#pragma unroll
    for (int e = 1; e < 8; ++e)
      if (r == e) d2 = acc[e];

    int nodeIdx = t * 16 + i;
    // dist < 1.0  <=>  dist^2 < 1.0 (both non-negative)
    if (active && d2 < 1.0f && nodeIdx <= nn) {
      out_adj[((size_t)b * N + nn) * N + nodeIdx] = 1.0f;  // row nn
      out_adj[((size_t)b * N + nodeIdx) * N + nn] = 1.0f;  // col nn
    }
  }
}

extern "C" void kernel_launch(void* const* d_in, const int* in_sizes, int n_in,
                              void* d_out, int out_size, void* d_ws, size_t ws_size,
                              hipStream_t stream) {
  const float* nodes = (const float*)d_in[0];
  const float* adj = (const float*)d_in[1];
  const float* ew = (const float*)d_in[2];
  const long long* num_nodes = (const long long*)d_in[3];  // int64 (B,1)

  int B = in_sizes[3];                                   // 32
  long adj_elems = in_sizes[1];                          // B*N*N
  int N = (int)(sqrt((double)(adj_elems / B)) + 0.5);    // 2048
  int F = (int)(in_sizes[0] / ((long)B * N));            // 64

  float* out_adj = (float*)d_out;
  float* out_ew = out_adj + (size_t)B * N * N;

  long n4 = (long)B * N * N / 4;  // float4 count per matrix
  // Pass 1: stream both matrices to the output (fully overwrites d_out).
  stream_copy_kernel<<<8192, 256, 0, stream>>>((const f4*)adj, (f4*)out_adj, n4);
  stream_copy_kernel<<<8192, 256, 0, stream>>>((const f4*)ew, (f4*)out_ew, n4);
  // Pass 2: WMMA distance mask + row/col scatter (same stream => ordered).
  dist_scatter_kernel<<<B, 256, 0, stream>>>(nodes, num_nodes, out_adj, N, F);
}